// LigerFusedLinearCrossEntropyHelion_11922829213913
// MI455X (gfx1250) — compile-verified
//
#include <hip/hip_runtime.h>

typedef __attribute__((ext_vector_type(2))) float v2f;
typedef __attribute__((ext_vector_type(4))) float v4f;
typedef __attribute__((ext_vector_type(8))) float v8f;
typedef __attribute__((ext_vector_type(4))) unsigned int v4u;
typedef __attribute__((ext_vector_type(8))) int v8i;
typedef __attribute__((ext_vector_type(4))) int v4i;

#define BT 4096
#define HD 2048
#define VO 32000
#define ROWS 16            // rows of x per block
#define COLS 256           // vocab columns per block iteration (8 waves * 2 tiles * 16)
#define LDS_STRIDE 2052    // 2048 + 4 pad -> conflict-free A reads (4*stride % 64banks = 4)

__global__ void flce_init_kernel(float* out, int* cnt) {
    out[0] = 0.0f;
    cnt[0] = 0;
}

__global__ void flce_count_kernel(const long long* __restrict__ tgt, int* cnt) {
    int i = blockIdx.x * blockDim.x + threadIdx.x;
    if (i < BT && tgt[i] != -100) atomicAdd(cnt, 1);
}

__global__ __launch_bounds__(256)
void flce_main_kernel(const float* __restrict__ x, const float* __restrict__ w,
                      const long long* __restrict__ tgt, const int* __restrict__ cnt,
                      float* __restrict__ out) {
    __shared__ float xs[ROWS * LDS_STRIDE];
    __shared__ float redM[8][16];
    __shared__ float redS[8][16];
    __shared__ float redP[8][16];

    const int tid  = threadIdx.x;
    const int wave = tid >> 5;       // 0..7
    const int lane = tid & 31;       // wave32
    const int half = lane >> 4;      // 0: K=0..1 / rows 0..7   1: K=2..3 / rows 8..15
    const int l16  = lane & 15;      // A: M index, B: N index
    const int koff = half * 2;
    const int row0 = blockIdx.x * ROWS;

    // ---- Stage the 16 x 2048 f32 x-tile into LDS via the Tensor Data Mover ----
    // One iterated 1x2048-elem tile per row: global advances 2048 elems/iter,
    // LDS advances 2052 elems/iter (reproduces the bank-conflict padding).
    if (wave == 0) {
        unsigned ldsAddr = (unsigned)(size_t)(void*)xs;               // addr[31:0] = LDS offset
        unsigned long long ga = (unsigned long long)(size_t)(x + (size_t)row0 * HD);
        v4u g0;
        g0[0] = 1u;                                                   // count=1, user D#
        g0[1] = ldsAddr;                                              // lds_addr
        g0[2] = (unsigned)(ga & 0xFFFFFFFFu);                         // global_addr[31:0]
        g0[3] = (unsigned)((ga >> 32) & 0x01FFFFFFu) | (2u << 30);    // global_addr[56:32] | type=2
        v8i g1;
        g1[0] = (2 << 16) | (1 << 19);        // data_size=4B, iterate_enable=1
        g1[1] = (int)((HD & 0xFFFFu) << 16);  // tensor_dim0[15:0]=2048
        g1[2] = (int)(ROWS << 16);            // tensor_dim0[31:16]=0 | tensor_dim1[15:0]=16
        g1[3] = (int)((unsigned)HD << 16);    // tensor_dim1[31:16]=0 | tile_dim0=2048
        g1[4] = 1;                            // tile_dim1=1, tile_dim2=0
        g1[5] = HD;                           // tensor_dim0_stride[31:0]=2048
        g1[6] = (int)((unsigned)HD << 16);    // stride0[47:32]=0 | tensor_dim1_stride[15:0]=2048
        g1[7] = 0;                            // tensor_dim1_stride[47:16]=0
        v4i g2;
        g2[0] = 1;                            // tensor_dim2 (unused)
        g2[1] = LDS_STRIDE;                   // lds_addr_increment per iteration
        g2[2] = HD;                           // global_addr_increment[31:0]
        g2[3] = (int)(15u << 16);             // increment[47:32]=0 | iterate_count=15 (=16x)
        v4i g3 = {0, 0, 0, 0};
        v8i g4 = {0, 0, 0, 0, 0, 0, 0, 0};    // extra group required by 6-arg builtin
        __builtin_amdgcn_tensor_load_to_lds(g0, g1, g2, g3, g4, 0);
        __builtin_amdgcn_s_wait_tensorcnt(0);
    }
    __syncthreads();

    // targets for the 8 rows this lane's C VGPRs cover (rows r + 8*half)
    int t[8];
    #pragma unroll
    for (int r = 0; r < 8; ++r) t[r] = (int)tgt[row0 + r + 8 * half];

    // running online-logsumexp state per C row
    float m[8], s[8], p[8];
    #pragma unroll
    for (int r = 0; r < 8; ++r) { m[r] = -__builtin_inff(); s[r] = 0.0f; p[r] = 0.0f; }

    const float* xb = &xs[l16 * LDS_STRIDE + koff];   // A feed from LDS (shared by both tiles)

    for (int vb = 0; vb < VO; vb += COLS) {
        const int col = vb + wave * 32 + l16;          // this lane's N column, tile 0
        const float* wb0 = w + (size_t)col * HD + koff;
        const float* wb1 = wb0 + (size_t)16 * HD;      // tile 1: +16 columns

        // prime L2 one vocab tile ahead (global_prefetch_b8)
        if (vb + COLS < VO) {
            __builtin_prefetch(wb0 + (size_t)COLS * HD, 0, 1);
            __builtin_prefetch(wb1 + (size_t)COLS * HD, 0, 1);
        }

        v8f c0 = {};
        v8f c1 = {};
        #pragma unroll 8
        for (int h = 0; h < HD; h += 4) {
            v2f a  = *(const v2f*)(xb + h);            // ds_load_b64, shared A
            v2f b0 = *(const v2f*)(wb0 + h);           // global_load_b64
            v2f b1 = *(const v2f*)(wb1 + h);
            c0 = __builtin_amdgcn_wmma_f32_16x16x4_f32(
                     false, a, false, b0, (short)0, c0, false, false);
            c1 = __builtin_amdgcn_wmma_f32_16x16x4_f32(
                     false, a, false, b1, (short)0, c1, false, false);
        }

        // online softmax update: merge both tiles' columns in one rescale
        #pragma unroll
        for (int r = 0; r < 8; ++r) {
            float v0 = c0[r];
            float v1 = c1[r];
            float M  = fmaxf(m[r], fmaxf(v0, v1));
            s[r] = s[r] * __expf(m[r] - M) + __expf(v0 - M) + __expf(v1 - M);
            m[r] = M;
            p[r] += (t[r] == col)      ? v0 : 0.0f;
            p[r] += (t[r] == col + 16) ? v1 : 0.0f;
        }
    }

    // ---- merge the 16 N-lanes (xor 8,4,2,1 stays inside each 16-lane half) ----
    #pragma unroll
    for (int r = 0; r < 8; ++r) {
        #pragma unroll
        for (int off = 8; off >= 1; off >>= 1) {
            float om = __shfl_xor(m[r], off);
            float os = __shfl_xor(s[r], off);
            float op = __shfl_xor(p[r], off);
            float M  = fmaxf(m[r], om);
            s[r] = s[r] * __expf(m[r] - M) + os * __expf(om - M);
            m[r] = M;
            p[r] += op;
        }
    }
    if (l16 == 0) {
        #pragma unroll
        for (int r = 0; r < 8; ++r) {
            redM[wave][r + 8 * half] = m[r];
            redS[wave][r + 8 * half] = s[r];
            redP[wave][r + 8 * half] = p[r];
        }
    }
    __syncthreads();

    // ---- merge the 8 waves, finish row loss, atomically accumulate ----
    if (tid < 16) {
        float M = -__builtin_inff(), S = 0.0f, P = 0.0f;
        #pragma unroll
        for (int wv = 0; wv < 8; ++wv) {
            float om = redM[wv][tid], os = redS[wv][tid];
            float Mn = fmaxf(M, om);
            S = S * __expf(M - Mn) + os * __expf(om - Mn);
            M = Mn;
            P += redP[wv][tid];
        }
        float lse  = M + __logf(S);
        float invN = 1.0f / (float)cnt[0];
        atomicAdd(out, (lse - P) * invN);
    }
}

extern "C" void kernel_launch(void* const* d_in, const int* in_sizes, int n_in,
                              void* d_out, int out_size, void* d_ws, size_t ws_size,
                              hipStream_t stream) {
    (void)in_sizes; (void)n_in; (void)out_size; (void)ws_size;
    const float*     x   = (const float*)d_in[0];
    const float*     w   = (const float*)d_in[1];
    const long long* tgt = (const long long*)d_in[2];
    float* out = (float*)d_out;
    int*   cnt = (int*)d_ws;

    flce_init_kernel<<<1, 1, 0, stream>>>(out, cnt);
    flce_count_kernel<<<BT / 256, 256, 0, stream>>>(tgt, cnt);
    flce_main_kernel<<<BT / ROWS, 256, 0, stream>>>(x, w, tgt, cnt, out);
}